// ReactorNet_36498632081790
// MI455X (gfx1250) — compile-verified
//
#include <hip/hip_runtime.h>
#include <hip/hip_bf16.h>
#include <math.h>

#define R_GASC 8.314462618f
#define P_ATMC 101325.0f
#define NS 53          // species
#define SP 56          // species padded (mult of 4, K dim of GEMM1)
#define LDSW 58        // LDS row stride (58 = 2*29 -> conflict-free, 8B aligned)
#define NR 325         // reactions
#define RP 336         // reactions padded (21 tiles of 16)
#define NT 21          // reaction N-tiles
#define K1 14          // GEMM1 K-steps (56/4)
#define OUTC 54        // output row: [dTdt, dYdt x 53]
#define WAVES 2        // waves per block
#define MROWS 32       // batch rows per wave (2 C-tiles)

typedef __attribute__((ext_vector_type(2))) float v2f;
typedef __attribute__((ext_vector_type(8))) float v8f;

// ---- workspace layout (float offsets), all fragment-packed for b64 loads ----
// GEMM1 B frags: index ((k4*NT + nt)*64 + lane*2 + comp), k4 in [0,K1)
#define PK1SZ (K1 * NT * 64)
#define OFF_PKR 0
#define OFF_PKP (OFF_PKR + PK1SZ)
#define OFF_PKN (OFF_PKP + PK1SZ)
// GEMM2 B frags: index ((k4*4 + st)*64 + lane*2 + comp), k4 in [0,84)
#define PK2SZ (84 * 4 * 64)
#define OFF_PKB (OFF_PKN + PK1SZ)
// per-reaction constants float4 {logA, b, Ea, dnu}, padded to RP
#define OFF_NC (OFF_PKB + PK2SZ)

__device__ __forceinline__ v8f vzero8() {
    v8f z;
#pragma unroll
    for (int i = 0; i < 8; ++i) z[i] = 0.0f;
    return z;
}

__device__ __forceinline__ v8f wmma4(v2f a, v2f b, v8f c) {
    // D = A(16x4 f32) x B(4x16 f32) + C(16x16 f32)
    return __builtin_amdgcn_wmma_f32_16x16x4_f32(
        false, a, false, b, (short)0, c, false, false);
}

// ---------------- prep: pre-swizzle stoichiometry into fragment layout ----------------
__global__ void reactor_prep(const float* __restrict__ nuR, const float* __restrict__ nuP,
                             const float* __restrict__ arrA, const float* __restrict__ arrB,
                             const float* __restrict__ arrEa, float* __restrict__ ws) {
    const int idx = blockIdx.x * blockDim.x + threadIdx.x;
    if (idx >= PK2SZ) return;

    // GEMM2 fragment pack: B = nu_net [K=RP][N=64]
    {
        const int k4 = idx >> 8;           // /256
        const int rem = idx & 255;
        const int st = rem >> 6;
        const int t = rem & 63;
        const int lane = t >> 1;
        const int comp = t & 1;
        const int K = k4 * 4 + ((lane >> 4) << 1) + comp;
        const int s = st * 16 + (lane & 15);
        float v = 0.f;
        if (K < NR && s < NS) v = nuP[K * NS + s] - nuR[K * NS + s];
        ws[OFF_PKB + idx] = v;
    }
    // GEMM1 fragment packs: B = nu_X^T [K=SP][N=RP]
    if (idx < PK1SZ) {
        const int k4 = idx / (NT * 64);
        const int rem = idx % (NT * 64);
        const int nt = rem >> 6;
        const int t = rem & 63;
        const int lane = t >> 1;
        const int comp = t & 1;
        const int K = k4 * 4 + ((lane >> 4) << 1) + comp;  // species index
        const int n = nt * 16 + (lane & 15);               // reaction index
        float nr = 0.f, np_ = 0.f;
        if (n < NR && K < NS) { nr = nuR[n * NS + K]; np_ = nuP[n * NS + K]; }
        ws[OFF_PKR + idx] = nr;
        ws[OFF_PKP + idx] = np_;
        ws[OFF_PKN + idx] = np_ - nr;
    }
    // per-reaction constants
    if (idx < RP) {
        const int r = idx;
        float d = 0.f;
        if (r < NR) {
            for (int s = 0; s < NS; ++s) d += nuP[r * NS + s] - nuR[r * NS + s];
        }
        ws[OFF_NC + 4 * r + 0] = (r < NR) ? __logf(arrA[r]) : -1e30f;  // pad -> exp()==0
        ws[OFF_NC + 4 * r + 1] = (r < NR) ? arrB[r] : 0.f;
        ws[OFF_NC + 4 * r + 2] = (r < NR) ? arrEa[r] : 0.f;
        ws[OFF_NC + 4 * r + 3] = d;
    }
}

// ---------------- fused reactor RHS ----------------
__global__ __launch_bounds__(32 * WAVES) void reactor_main(
    const float* __restrict__ Tarr, const float* __restrict__ Parr,
    const float* __restrict__ Yarr,
    const float* __restrict__ nasaL, const float* __restrict__ nasaH,
    const float* __restrict__ Tmid, const float* __restrict__ Wmol,
    const float* __restrict__ ws, float* __restrict__ out, int B) {
    __shared__ float s_logC[WAVES][MROWS][LDSW];
    __shared__ float s_g[WAVES][MROWS][LDSW];
    __shared__ float s_h[WAVES][MROWS][LDSW];
    __shared__ float s_q[WAVES][MROWS][18];
    __shared__ float s_T[WAVES][MROWS], s_logT[WAVES][MROWS], s_rcpRT[WAVES][MROWS];
    __shared__ float s_lcs[WAVES][MROWS], s_rcprho[WAVES][MROWS];
    __shared__ float s_cpm[WAVES][MROWS], s_dtn[WAVES][MROWS];

    const int w = threadIdx.x >> 5;
    const int lane = threadIdx.x & 31;
    const int half = lane >> 4;
    const int l15 = lane & 15;
    const int b0 = (blockIdx.x * WAVES + w) * MROWS;

    const float Pv = Parr[0];

    // ---- phase 0+1: one batch row per lane: scalars + thermo tiles ----
    {
        const int brow = min(b0 + lane, B - 1);
        float invmW = 0.f;
        for (int s = 0; s < NS; ++s) invmW += Yarr[brow * NS + s] / Wmol[s];
        const float Tv = Tarr[brow];
        const float lT = __logf(Tv);
        const float rho = Pv / (R_GASC * Tv * invmW);
        s_T[w][lane] = Tv;
        s_logT[w][lane] = lT;
        s_rcpRT[w][lane] = 1.f / (R_GASC * Tv);
        s_lcs[w][lane] = __logf(P_ATMC / R_GASC) - lT;
        s_rcprho[w][lane] = 1.f / rho;

        const float T2 = Tv * Tv, T3 = T2 * Tv, T4 = T2 * T2, rT = 1.f / Tv;
        float cpacc = 0.f;
        for (int s = 0; s < SP; ++s) {
            float lC = 0.f, gv = 0.f, hv = 0.f;
            if (s < NS) {
                float Ws = Wmol[s];
                float Yv = Yarr[brow * NS + s];
                float C = rho * Yv / Ws;
                lC = __logf(fmaxf(C, 1e-30f));
                const float* c = (Tv < Tmid[s]) ? (nasaL + s * 7) : (nasaH + s * 7);
                float a0 = c[0], a1 = c[1], a2 = c[2], a3 = c[3], a4 = c[4], a5 = c[5],
                      a6 = c[6];
                float cp = a0 + a1 * Tv + a2 * T2 + a3 * T3 + a4 * T4;
                hv = a0 + 0.5f * a1 * Tv + (1.f / 3.f) * a2 * T2 + 0.25f * a3 * T3 +
                     0.2f * a4 * T4 + a5 * rT;
                float sE = a0 * lT + a1 * Tv + 0.5f * a2 * T2 + (1.f / 3.f) * a3 * T3 +
                           0.25f * a4 * T4 + a6;
                gv = hv - sE;
                cpacc += Yv * cp * (R_GASC / Ws);
            }
            s_logC[w][lane][s] = lC;
            s_g[w][lane][s] = gv;
            s_h[w][lane][s] = hv;
        }
        s_cpm[w][lane] = cpacc;
    }
    __syncthreads();

    const float* pkR = ws + OFF_PKR;
    const float* pkP = ws + OFF_PKP;
    const float* pkN = ws + OFF_PKN;
    const float* pkB = ws + OFF_PKB;

    // ---- phase 2+3: per reaction N-tile: 3x2 WMMA GEMMs -> q -> wdot GEMM ----
    v8f acc2[2][4];
#pragma unroll
    for (int rt = 0; rt < 2; ++rt)
#pragma unroll
        for (int st = 0; st < 4; ++st) acc2[rt][st] = vzero8();

    for (int nt = 0; nt < NT; ++nt) {
        const float4 nc = *(const float4*)(ws + OFF_NC + 4 * (nt * 16 + l15));
        const float la = nc.x, bb = nc.y, ea = nc.z, dnu = nc.w;

        v8f aR[2] = {vzero8(), vzero8()};
        v8f aP[2] = {vzero8(), vzero8()};
        v8f aG[2] = {vzero8(), vzero8()};

#pragma unroll 2
        for (int k4 = 0; k4 < K1; ++k4) {
            const int kx = k4 * 4 + 2 * half;
            const v2f aC0 = *(const v2f*)&s_logC[w][l15][kx];
            const v2f aC1 = *(const v2f*)&s_logC[w][16 + l15][kx];
            const v2f aG0 = *(const v2f*)&s_g[w][l15][kx];
            const v2f aG1 = *(const v2f*)&s_g[w][16 + l15][kx];
            const int base = (k4 * NT + nt) * 64 + lane * 2;
            const v2f bR = *(const v2f*)(pkR + base);
            const v2f bP = *(const v2f*)(pkP + base);
            const v2f bN = *(const v2f*)(pkN + base);
            aR[0] = wmma4(aC0, bR, aR[0]);
            aR[1] = wmma4(aC1, bR, aR[1]);
            aP[0] = wmma4(aC0, bP, aP[0]);
            aP[1] = wmma4(aC1, bP, aP[1]);
            aG[0] = wmma4(aG0, bN, aG[0]);
            aG[1] = wmma4(aG1, bN, aG[1]);
        }

        // q = exp(log_kf + A.nu_r) - exp(log_kf + dg - dnu*log_cstd + A.nu_p)
        float qv[2][8];
#pragma unroll
        for (int rt = 0; rt < 2; ++rt) {
#pragma unroll
            for (int j = 0; j < 8; ++j) {
                const int row = rt * 16 + j + 8 * half;
                float lkf = la + bb * s_logT[w][row] - ea * s_rcpRT[w][row];
                float qf = __expf(lkf + aR[rt][j]);
                float qr = __expf(lkf + aG[rt][j] - dnu * s_lcs[w][row] + aP[rt][j]);
                qv[rt][j] = qf - qr;
            }
        }
        __syncthreads();
#pragma unroll
        for (int rt = 0; rt < 2; ++rt)
#pragma unroll
            for (int j = 0; j < 8; ++j) s_q[w][rt * 16 + j + 8 * half][l15] = qv[rt][j];
        __syncthreads();

        // wdot partial: q_tile (K = these 16 reactions) x nu_net
#pragma unroll
        for (int kk = 0; kk < 16; kk += 4) {
            const int kxx = kk + 2 * half;
            const v2f a20 = *(const v2f*)&s_q[w][l15][kxx];
            const v2f a21 = *(const v2f*)&s_q[w][16 + l15][kxx];
            const int k4g = nt * 4 + (kk >> 2);
#pragma unroll
            for (int st = 0; st < 4; ++st) {
                const v2f b2 = *(const v2f*)(pkB + (k4g * 4 + st) * 64 + lane * 2);
                acc2[0][st] = wmma4(a20, b2, acc2[0][st]);
                acc2[1][st] = wmma4(a21, b2, acc2[1][st]);
            }
        }
    }

    // ---- phase 4: dY/dt writes + dT/dt reduction ----
#pragma unroll
    for (int rt = 0; rt < 2; ++rt) {
#pragma unroll
        for (int j = 0; j < 8; ++j) {
            const int row = rt * 16 + j + 8 * half;
            const int brow2 = b0 + row;
            const bool ok = brow2 < B;
            const float rcpr = s_rcprho[w][row];
            float hpart = 0.f;
#pragma unroll
            for (int st = 0; st < 4; ++st) {
                const int s = st * 16 + l15;
                const float wd = acc2[rt][st][j];
                if (s < NS) {
                    hpart += s_h[w][row][s] * wd;
                    if (ok) out[brow2 * OUTC + 1 + s] = wd * Wmol[s] * rcpr;
                }
            }
#pragma unroll
            for (int msk = 1; msk < 16; msk <<= 1) hpart += __shfl_xor(hpart, msk, 32);
            if (l15 == 0) s_dtn[w][row] = hpart;
        }
    }
    __syncthreads();
    if ((b0 + lane) < B) {
        const float num = s_dtn[w][lane];
        const float dT =
            -num * R_GASC * s_T[w][lane] * s_rcprho[w][lane] / s_cpm[w][lane];
        out[(b0 + lane) * OUTC] = dT;
    }
}

extern "C" void kernel_launch(void* const* d_in, const int* in_sizes, int n_in,
                              void* d_out, int out_size, void* d_ws, size_t ws_size,
                              hipStream_t stream) {
    // setup_inputs order: t, T, P, Y, nasa_low, nasa_high, nasa_T_mid,
    //                     mol_weights, arr_A, arr_b, arr_Ea, nu_react, nu_prod
    const float* T = (const float*)d_in[1];
    const float* P = (const float*)d_in[2];
    const float* Y = (const float*)d_in[3];
    const float* nL = (const float*)d_in[4];
    const float* nH = (const float*)d_in[5];
    const float* Tm = (const float*)d_in[6];
    const float* Wm = (const float*)d_in[7];
    const float* A = (const float*)d_in[8];
    const float* bb = (const float*)d_in[9];
    const float* Ea = (const float*)d_in[10];
    const float* nR = (const float*)d_in[11];
    const float* nP = (const float*)d_in[12];
    float* out = (float*)d_out;
    float* ws = (float*)d_ws;

    const int B = in_sizes[1];

    reactor_prep<<<(PK2SZ + 127) / 128, 128, 0, stream>>>(nR, nP, A, bb, Ea, ws);

    const int tiles = (B + MROWS - 1) / MROWS;
    const int blocks = (tiles + WAVES - 1) / WAVES;
    reactor_main<<<blocks, 32 * WAVES, 0, stream>>>(T, P, Y, nL, nH, Tm, Wm, ws, out, B);
}